// Model_6253472383143
// MI455X (gfx1250) — compile-verified
//
#include <hip/hip_runtime.h>

typedef __attribute__((ext_vector_type(2))) float v2f;
typedef __attribute__((ext_vector_type(8))) float v8f;

#define EMB_DIM 32
#define LHIST   10240   // LDS histogram capacity (40 KB of the 320 KB WGP LDS)

// ---------------------------------------------------------------- zero
__global__ void zero_u32_kernel(unsigned int* __restrict__ p, int n) {
    int i = blockIdx.x * blockDim.x + threadIdx.x;
    if (i < n) p[i] = 0u;
}

// ---------------------------------------------------------------- histogram
__global__ void hist_kernel(const int* __restrict__ hist_src, int hist_len,
                            unsigned int* __restrict__ gcounts, int num_items) {
    __shared__ unsigned int lh[LHIST];
    const int stride = gridDim.x * blockDim.x;
    const int gtid   = blockIdx.x * blockDim.x + threadIdx.x;

    if (num_items <= LHIST) {
        for (int i = threadIdx.x; i < num_items; i += blockDim.x) lh[i] = 0u;
        __syncthreads();

        const int nvec = hist_len >> 2;                  // int4-vectorized body
        const int4* src4 = (const int4*)hist_src;
        for (int i = gtid; i < nvec; i += stride) {
            __builtin_prefetch(&src4[i + stride], 0, 0); // global_prefetch_b8
            int4 v = src4[i];
            atomicAdd(&lh[v.x], 1u);                     // ds_add_u32
            atomicAdd(&lh[v.y], 1u);
            atomicAdd(&lh[v.z], 1u);
            atomicAdd(&lh[v.w], 1u);
        }
        for (int i = (nvec << 2) + gtid; i < hist_len; i += stride)
            atomicAdd(&lh[hist_src[i]], 1u);             // tail

        __syncthreads();
        for (int i = threadIdx.x; i < num_items; i += blockDim.x) {
            unsigned int c = lh[i];
            if (c) atomicAdd(&gcounts[i], c);            // global_atomic_add_u32
        }
    } else {
        // fallback: direct global atomics (not expected for this problem size)
        for (int i = gtid; i < hist_len; i += stride)
            atomicAdd(&gcounts[hist_src[i]], 1u);
    }
}

// ---------------------------------------------------------------- weighted mean-pool
// u[d] = (1/H) * sum_i counts[i] * E[i][d]
__global__ void pool_kernel(const unsigned int* __restrict__ counts,
                            const float* __restrict__ emb,
                            int num_items, float inv_h,
                            float* __restrict__ u) {
    __shared__ float part[8][EMB_DIM];
    const int d = threadIdx.x & (EMB_DIM - 1);   // dim 0..31
    const int g = threadIdx.x >> 5;              // group 0..7 (one wave32 each)
    float acc = 0.0f;
    for (int i = g; i < num_items; i += 8) {
        float c = (float)counts[i];              // counts < 2^24, exact in f32
        acc += c * emb[(size_t)i * EMB_DIM + d];
    }
    part[g][d] = acc;
    __syncthreads();
    if (g == 0) {
        float s = 0.0f;
        #pragma unroll
        for (int j = 0; j < 8; ++j) s += part[j][d];
        u[d] = s * inv_h;
    }
}

// ---------------------------------------------------------------- WMMA scoring
// One wave scores a 16-item tile:  D = A(16x4, u replicated per row) x B(4x16, E^T tile) + C
// accumulated over K=32 in 8 steps of V_WMMA_F32_16X16X4_F32.
//
// f32 A 16x4 layout : lanes 0-15 -> {K=0,K=1}, lanes 16-31 -> {K=2,K=3}  (2 VGPRs)
// f32 B 4x16 layout : VGPR0/1 lanes 0-15 -> rows K=0/1, lanes 16-31 -> rows K=2/3
// f32 D 16x16       : VGPR0 lanes 0-15 = row M=0, N=lane  -> the 16 scores
__global__ void score_wmma_kernel(const float* __restrict__ emb,
                                  const float* __restrict__ u,
                                  float* __restrict__ scores, int num_items) {
    const int lane = threadIdx.x & 31;
    const int wave = threadIdx.x >> 5;
    const int tile = blockIdx.x * (blockDim.x >> 5) + wave;
    const int ntiles = (num_items + 15) >> 4;
    if (tile >= ntiles) return;                  // wave-uniform: EXEC stays all-1s

    const int n  = tile * 16 + (lane & 15);
    const int nc = (n < num_items) ? n : (num_items - 1);  // clamp, keep EXEC full
    const int kk = (lane >> 4) << 1;             // 0 for lanes 0-15, 2 for lanes 16-31
    const float* __restrict__ erow = emb + (size_t)nc * EMB_DIM;

    v8f c = {};
    #pragma unroll
    for (int k0 = 0; k0 < EMB_DIM; k0 += 4) {
        v2f a, b;
        a.x = u[k0 + kk];
        a.y = u[k0 + kk + 1];
        b.x = erow[k0 + kk];
        b.y = erow[k0 + kk + 1];
        // (neg_a, A, neg_b, B, c_mod, C, reuse_a, reuse_b)
        c = __builtin_amdgcn_wmma_f32_16x16x4_f32(false, a, false, b,
                                                  (short)0, c, false, false);
    }
    if (lane < 16 && n < num_items) scores[n] = c[0];
}

// ---------------------------------------------------------------- top-k
// k sequential arg-max passes; key = (ordered float bits << 32) | ~index
// -> max score first, then lowest index (matches jax.lax.top_k tie order).
__global__ void topk_kernel(const float* __restrict__ scores, int num_items,
                            const int* __restrict__ kptr, int* __restrict__ out) {
    __shared__ unsigned long long best;
    __shared__ int sel[64];
    int k = *kptr;
    if (k > 64) k = 64;
    if (k < 0)  k = 0;

    for (int p = 0; p < k; ++p) {
        if (threadIdx.x == 0) best = 0ull;
        __syncthreads();

        unsigned long long my = 0ull;
        for (int i = threadIdx.x; i < num_items; i += blockDim.x) {
            bool skip = false;
            for (int j = 0; j < p; ++j)
                if (sel[j] == i) { skip = true; break; }
            if (skip) continue;
            unsigned int ub = __float_as_uint(scores[i]);
            unsigned int key32 = (ub & 0x80000000u) ? ~ub : (ub | 0x80000000u);
            unsigned long long key =
                ((unsigned long long)key32 << 32) |
                (unsigned long long)(0xFFFFFFFFu - (unsigned int)i);
            if (key > my) my = key;
        }
        atomicMax(&best, my);
        __syncthreads();

        if (threadIdx.x == 0) {
            int idx = (int)(0xFFFFFFFFu - (unsigned int)(best & 0xFFFFFFFFull));
            sel[p] = idx;
            out[p] = idx;
        }
        __syncthreads();
    }
}

// ---------------------------------------------------------------- launcher
extern "C" void kernel_launch(void* const* d_in, const int* in_sizes, int n_in,
                              void* d_out, int out_size, void* d_ws, size_t ws_size,
                              hipStream_t stream) {
    const int*   hist = (const int*)d_in[0];     // user_history  [H] int32
    const float* emb  = (const float*)d_in[1];   // item_embeddings [N,32] f32
    const int*   kptr = (const int*)d_in[2];     // k (scalar on device)

    const int hist_len  = in_sizes[0];
    const int num_items = in_sizes[1] / EMB_DIM;

    // workspace layout: [0,40KB) counts u32 | [64KB) u[32] f32 | [68KB) scores f32
    unsigned int* counts = (unsigned int*)d_ws;
    float* u      = (float*)((char*)d_ws + 64 * 1024);
    float* scores = (float*)((char*)d_ws + 68 * 1024);
    int*   out    = (int*)d_out;

    zero_u32_kernel<<<(num_items + 255) / 256, 256, 0, stream>>>(counts, num_items);

    hist_kernel<<<1024, 256, 0, stream>>>(hist, hist_len, counts, num_items);

    pool_kernel<<<1, 256, 0, stream>>>(counts, emb, num_items,
                                       1.0f / (float)hist_len, u);

    const int ntiles = (num_items + 15) / 16;
    const int waves_per_block = 8;   // 256 threads = 8 wave32
    score_wmma_kernel<<<(ntiles + waves_per_block - 1) / waves_per_block,
                        waves_per_block * 32, 0, stream>>>(emb, u, scores, num_items);

    topk_kernel<<<1, 256, 0, stream>>>(scores, num_items, kptr, out);
}